// RelGAT_30399778521236
// MI455X (gfx1250) — compile-verified
//
#include <hip/hip_runtime.h>
#include <math.h>

// Problem constants (from reference setup_inputs)
#define B_  2
#define N_  2048
#define D_  512
#define F_  64
#define H_  4
#define R_  3
#define ALPHA_ 0.2f
#define NEGV_ -9e15f

// CDNA5 fp32 WMMA: D(16x16 f32, 8 VGPRs) = A(16x4 f32, 2 VGPRs) * B(4x16 f32, 2 VGPRs) + C
typedef __attribute__((ext_vector_type(2))) float v2f;
typedef __attribute__((ext_vector_type(8))) float v8f;

static __device__ __forceinline__ v8f wmma_f32(v2f a, v2f b, v8f c) {
    // 8 args: (neg_a, A, neg_b, B, c_mod, C, reuse_a, reuse_b)
    return __builtin_amdgcn_wmma_f32_16x16x4_f32(false, a, false, b, (short)0, c, false, false);
}

// ---------------------------------------------------------------------------
// Kernel 1: Wh[h,b,n,f] = x[b]·W[h], rWh[h,r,b,n,f] = x[b]·W_out[h,r]
// One wave per 16x16 output tile; K=512 in 128 WMMA steps.
// ---------------------------------------------------------------------------
__global__ __launch_bounds__(256)
void k_input_gemm(const float* __restrict__ x,
                  const float* __restrict__ W,
                  const float* __restrict__ Wout,
                  float* __restrict__ Wh,
                  float* __restrict__ rWh) {
    const int tid  = threadIdx.x;
    const int lane = tid & 31;
    const int wid  = tid >> 5;
    int g = blockIdx.x * 8 + wid;          // global wave id, 16384 total
    const int ct = g & 3;   g >>= 2;       // col tile (F/16)
    const int rt = g & 127; g >>= 7;       // row tile (N/16)
    const int b  = g & 1;   g >>= 1;
    const int s  = g & 3;   g >>= 2;       // slot: 0 -> W, 1..3 -> W_out[r]
    const int h  = g;                      // head

    const float* wp;
    float* op;
    if (s == 0) {
        wp = W + (size_t)h * D_ * F_;
        op = Wh + ((size_t)(h * B_ + b) * N_) * F_;
    } else {
        const int r = s - 1;
        wp = Wout + ((size_t)(h * R_ + r) * D_) * F_;
        op = rWh + ((size_t)((h * R_ + r) * B_ + b) * N_) * F_;
    }

    const int l15   = lane & 15;
    const int khalf = (lane >> 4) * 2;     // A/B K-half layout (lanes 16-31 hold K+2)
    const int arow  = rt * 16 + l15;
    const int bcol  = ct * 16 + l15;
    const float* xrow = x + ((size_t)b * N_ + arow) * D_;

    v8f acc = {};
    for (int k0 = 0; k0 < D_; k0 += 4) {
        v2f a, bb;
        a.x  = xrow[k0 + khalf];
        a.y  = xrow[k0 + khalf + 1];
        bb.x = wp[(size_t)(k0 + khalf)     * F_ + bcol];
        bb.y = wp[(size_t)(k0 + khalf + 1) * F_ + bcol];
        acc = wmma_f32(a, bb, acc);
    }

    const int mbase = rt * 16 + ((lane >> 4) << 3);   // C layout: M = vgpr + 8*(lane/16)
#pragma unroll
    for (int v = 0; v < 8; ++v)
        op[(size_t)(mbase + v) * F_ + bcol] = acc[v];
}

// ---------------------------------------------------------------------------
// Kernel 2: Wh1 = Wh·a[:F], Wh2 = Wh·a[F:]  (one thread per (h,b,n) row)
// ---------------------------------------------------------------------------
__global__ void k_attn_proj(const float* __restrict__ Wh,
                            const float* __restrict__ a,
                            float* __restrict__ Wh1,
                            float* __restrict__ Wh2) {
    const int idx = blockIdx.x * blockDim.x + threadIdx.x;   // [0, H*B*N)
    if (idx >= H_ * B_ * N_) return;
    const int h = idx / (B_ * N_);
    const float* row = Wh + (size_t)idx * F_;
    const float* ah  = a + (size_t)h * 2 * F_;
    float s1 = 0.f, s2 = 0.f;
#pragma unroll 8
    for (int f = 0; f < F_; ++f) {
        const float w = row[f];
        s1 += w * ah[f];
        s2 += w * ah[F_ + f];
    }
    Wh1[idx] = s1;
    Wh2[idx] = s2;
}

// ---------------------------------------------------------------------------
// Kernel 3: column softmax stats (softmax over axis=1, i.e. over i for each j).
// Online max/sum; lanes span j (coalesced adj reads), loop over i.
// Stores the RECIPROCAL of the sum so the hot kernel multiplies instead of
// dividing per element.
// ---------------------------------------------------------------------------
__global__ void k_softmax_stats(const float* __restrict__ adj,
                                const float* __restrict__ Wh1,
                                const float* __restrict__ Wh2,
                                float* __restrict__ colmax,
                                float* __restrict__ colsum_rcp) {
    const int j  = blockIdx.x * 256 + threadIdx.x;
    const int b  = blockIdx.y;
    const int h  = blockIdx.z;
    const int hb = (h * B_ + b) * N_;
    const float  w2  = Wh2[hb + j];
    const float* w1  = Wh1 + hb;
    const float* acol = adj + ((size_t)b * N_) * N_ + j;

    float m = -3.0e38f, s = 0.f;
    for (int i = 0; i < N_; ++i) {
        const float av = acol[(size_t)i * N_];
        float v = w1[i] + w2;
        v = v > 0.f ? v : ALPHA_ * v;                 // leaky_relu
        const float e = av > 0.f ? v : NEGV_;         // mask
        if (e > m) { s *= __expf(m - e); m = e; }     // online softmax update
        s += __expf(e - m);
    }
    colmax[hb + j]     = m;
    colsum_rcp[hb + j] = 1.0f / s;
}

// ---------------------------------------------------------------------------
// Kernel 4 (dominant, ~13 GFLOP): fused attention recompute + GEMM + ELU.
//   out[b,i,h*F+f] = elu( sum_j sum_r att(i,j)*rel_adj[r,b,i,j]*rWh[h,r,b,j,f] )
// Block: 64 rows x 64 cols for one (h,b); 8 waves = 2 row-groups x 4 col-tiles.
// Each wave holds TWO 16x16 accumulators sharing every B fragment, so each
// global B load feeds two WMMAs (2x better WMMA:VMEM ratio than 32-row tiles).
// A-tiles (att*rel) are built cooperatively in LDS each 16-wide K step.
// ---------------------------------------------------------------------------
__global__ __launch_bounds__(256)
void k_attn_gemm(const float* __restrict__ adj,
                 const float* __restrict__ rel_adj,
                 const float* __restrict__ rWh,
                 const float* __restrict__ Wh1,
                 const float* __restrict__ Wh2,
                 const float* __restrict__ colmax,
                 const float* __restrict__ colsum_rcp,
                 float* __restrict__ out) {
    __shared__ float aT[R_][64][17];   // +1 pad: A-frag reads stride 17 -> conflict-free

    const int ib  = blockIdx.x * 64;   // row base
    const int b   = blockIdx.y;
    const int h   = blockIdx.z;
    const int tid  = threadIdx.x;
    const int lane = tid & 31;
    const int wid  = tid >> 5;
    const int rg   = wid >> 2;         // row group 0..1 (owns rowtiles rg*2, rg*2+1)
    const int ctl  = wid & 3;          // col tile 0..3
    const int hb   = (h * B_ + b) * N_;
    const int l15  = lane & 15;
    const int khalf = (lane >> 4) * 2;

    // Fill-phase coordinates: each thread owns ONE j-column of the tile
    // (tid & 15 is invariant across the 4 fill iterations since 256 % 16 == 0),
    // so all per-column softmax state is loaded once per j-step.
    const int jl  = tid & 15;
    const int il0 = tid >> 4;          // row offsets il0 + 16*t, t = 0..3

    v8f acc0 = {}, acc1 = {};

    for (int jb = 0; jb < N_; jb += 16) {
        const int j     = jb + jl;
        const float w2  = Wh2[hb + j];
        const float cm  = colmax[hb + j];
        const float rcs = colsum_rcp[hb + j];

        __syncthreads();               // previous tile fully consumed
#pragma unroll
        for (int t = 0; t < 4; ++t) {  // 1024 (i,j) pairs / 256 threads
            const int il = il0 + t * 16;
            const int i  = ib + il;
            const float av = adj[((size_t)b * N_ + i) * N_ + j];
            float v = Wh1[hb + i] + w2;
            v = v > 0.f ? v : ALPHA_ * v;                       // leaky_relu
            const float e   = av > 0.f ? v : NEGV_;             // mask
            const float att = __expf(e - cm) * rcs * av;        // softmax * adj
#pragma unroll
            for (int r = 0; r < R_; ++r) {
                const float* rp = rel_adj + (((size_t)r * B_ + b) * N_ + i) * N_;
                aT[r][il][jl] = att * rp[j];
                if (jb + 16 < N_)                   // speculative prefetch, next j-tile
                    __builtin_prefetch(rp + j + 16, 0, 1);
            }
        }
        __syncthreads();

#pragma unroll
        for (int r = 0; r < R_; ++r) {
            const float* bp = rWh + ((size_t)((h * R_ + r) * B_ + b) * N_) * F_;
#pragma unroll
            for (int kk = 0; kk < 4; ++kk) {
                const int k  = kk * 4 + khalf;
                const int jg = jb + k;
                v2f bb, a0, a1;
                bb.x = bp[(size_t)jg       * F_ + ctl * 16 + l15];
                bb.y = bp[(size_t)(jg + 1) * F_ + ctl * 16 + l15];
                a0.x = aT[r][rg * 32      + l15][k];
                a0.y = aT[r][rg * 32      + l15][k + 1];
                a1.x = aT[r][rg * 32 + 16 + l15][k];
                a1.y = aT[r][rg * 32 + 16 + l15][k + 1];
                acc0 = wmma_f32(a0, bb, acc0);     // B fragment reused by both
                acc1 = wmma_f32(a1, bb, acc1);
            }
        }
    }

    // Fused ELU + concat-head store: out[b, i, h*F + f]
    const int col    = h * F_ + ctl * 16 + l15;
    const int rbase0 = ib + rg * 32 + ((lane >> 4) << 3);   // C layout: M = v + 8*(lane/16)
    const int rbase1 = rbase0 + 16;
#pragma unroll
    for (int v = 0; v < 8; ++v) {
        const float x0 = acc0[v];
        const float x1 = acc1[v];
        out[((size_t)b * N_ + rbase0 + v) * (H_ * F_) + col] =
            x0 > 0.f ? x0 : (__expf(x0) - 1.f);
        out[((size_t)b * N_ + rbase1 + v) * (H_ * F_) + col] =
            x1 > 0.f ? x1 : (__expf(x1) - 1.f);
    }
}

// ---------------------------------------------------------------------------
extern "C" void kernel_launch(void* const* d_in, const int* in_sizes, int n_in,
                              void* d_out, int out_size, void* d_ws, size_t ws_size,
                              hipStream_t stream) {
    const float* x    = (const float*)d_in[0];
    const float* adj  = (const float*)d_in[1];
    const float* rel  = (const float*)d_in[2];
    const float* W    = (const float*)d_in[3];
    const float* a    = (const float*)d_in[4];
    const float* Wout = (const float*)d_in[5];
    float* out = (float*)d_out;

    // workspace carve-up (floats): ~16.3 MB total
    float* ws   = (float*)d_ws;
    size_t o = 0;
    float* Wh   = ws + o; o += (size_t)H_ * B_ * N_ * F_;        // 1,048,576
    float* rWh  = ws + o; o += (size_t)H_ * R_ * B_ * N_ * F_;   // 3,145,728
    float* Wh1  = ws + o; o += (size_t)H_ * B_ * N_;
    float* Wh2  = ws + o; o += (size_t)H_ * B_ * N_;
    float* cmax = ws + o; o += (size_t)H_ * B_ * N_;
    float* csum = ws + o; o += (size_t)H_ * B_ * N_;
    (void)ws_size; (void)in_sizes; (void)n_in; (void)out_size;

    // 1) input GEMMs: 16384 wave-tiles / 8 waves per block
    k_input_gemm<<<2048, 256, 0, stream>>>(x, W, Wout, Wh, rWh);
    // 2) attention projections
    k_attn_proj<<<(H_ * B_ * N_ + 255) / 256, 256, 0, stream>>>(Wh, a, Wh1, Wh2);
    // 3) column softmax stats (stores reciprocal sum)
    dim3 g3(N_ / 256, B_, H_);
    k_softmax_stats<<<g3, 256, 0, stream>>>(adj, Wh1, Wh2, cmax, csum);
    // 4) fused attention GEMM + ELU, 64-row blocks
    dim3 g4(N_ / 64, B_, H_);
    k_attn_gemm<<<g4, 256, 0, stream>>>(adj, rel, rWh, Wh1, Wh2, cmax, csum, out);
}